// SimpleGCNLayer_9775345566346
// MI455X (gfx1250) — compile-verified
//
#include <hip/hip_runtime.h>
#include <stdint.h>

// Problem constants (match reference)
#define NN 16384
#define DIN 128
#define DOUT 128
#define NEDGES 524288
#define ROW_WORDS 512                 // 16384 bits / 32
#define MASK_WORDS ((size_t)NN * ROW_WORDS)       // 8,388,608 u32 = 32 MB
#define MASK_BYTES (MASK_WORDS * 4)
#define DIS_BYTES ((size_t)NN * 4)
#define XS_BYTES ((size_t)NN * DIN * 4)

typedef __attribute__((ext_vector_type(2))) float v2f;
typedef __attribute__((ext_vector_type(8))) float v8f;

// ---------------- 1. zero the adjacency bit-mask (vectorized stores) --------
__global__ __launch_bounds__(256) void k_zero_mask(uint4* __restrict__ m) {
    size_t i = (size_t)blockIdx.x * blockDim.x + threadIdx.x;   // 2,097,152 uint4
    uint4 z; z.x = 0u; z.y = 0u; z.z = 0u; z.w = 0u;
    m[i] = z;
}

// ---------------- 2. set diagonal bits (fill_diagonal_(1.0)) ----------------
// diag bit positions are 16385 apart (>32), so each touched word is unique ->
// plain store is race-free after the zero pass.
__global__ __launch_bounds__(256) void k_diag(uint32_t* __restrict__ mask) {
    int r = blockIdx.x * blockDim.x + threadIdx.x;              // 16384 threads
    size_t bp = (size_t)r * (NN + 1);
    mask[bp >> 5] = 1u << (bp & 31u);
}

// ---------------- 3. scatter edges into mask (dedupe via atomicOr) ----------
__global__ __launch_bounds__(256) void k_edges(const int* __restrict__ ei,
                                               uint32_t* __restrict__ mask) {
    int e = blockIdx.x * blockDim.x + threadIdx.x;              // 524288 threads
    uint32_t r = (uint32_t)ei[e];
    uint32_t c = (uint32_t)ei[NEDGES + e];
    size_t bp = (size_t)r * NN + c;
    atomicOr(&mask[bp >> 5], 1u << (bp & 31u));
}

// ---------------- 4. degree -> dis = deg^-1/2 (row popcount) ----------------
__global__ __launch_bounds__(128) void k_degree(const uint32_t* __restrict__ mask,
                                                float* __restrict__ dis) {
    int r = blockIdx.x;
    int t = threadIdx.x;
    const uint32_t* row = mask + (size_t)r * ROW_WORDS;
    int cnt = 0;
    #pragma unroll
    for (int i = 0; i < 4; ++i) cnt += __popc(row[t + i * 128]);
    __shared__ int red[128];
    red[t] = cnt;
    __syncthreads();
    for (int s = 64; s > 0; s >>= 1) {
        if (t < s) red[t] += red[t + s];
        __syncthreads();
    }
    if (t == 0) dis[r] = rsqrtf((float)red[0]);   // deg >= 1 (diag), clamp implicit
}

// ---------------- 5. xs[c][:] = dis[c] * x[c][:] ----------------------------
__global__ __launch_bounds__(256) void k_scale_x(const float4* __restrict__ x,
                                                 const float* __restrict__ dis,
                                                 float4* __restrict__ xs) {
    int i = blockIdx.x * blockDim.x + threadIdx.x;              // NN*32 float4s
    float s = dis[i >> 5];                                      // 32 float4 per row
    float4 v = x[i];
    v.x *= s; v.y *= s; v.z *= s; v.w *= s;
    xs[i] = v;
}

// ---------------- 6. row aggregation: h[r] = dis[r] * sum_c xs[c] -----------
// One 128-thread workgroup per row. Deterministic neighbor compaction:
// chunked mask scan + prefix placement into LDS, then every thread owns one
// of the 128 feature dims and gathers (broadcast LDS index, coalesced xs).
__global__ __launch_bounds__(128) void k_aggregate(const uint32_t* __restrict__ mask,
                                                   const float* __restrict__ xs,
                                                   const float* __restrict__ dis,
                                                   float* __restrict__ h) {
    int r = blockIdx.x;
    int t = threadIdx.x;        // also the feature dim this thread accumulates
    __shared__ unsigned short nbr[4096];   // worst case one chunk = 128*32 bits
    __shared__ int counts[128];
    __shared__ int offs[129];

    const uint32_t* row = mask + (size_t)r * ROW_WORDS;
    float acc = 0.0f;

    for (int chunk = 0; chunk < 4; ++chunk) {
        uint32_t wbits = row[chunk * 128 + t];
        counts[t] = __popc(wbits);
        __syncthreads();
        if (t == 0) {
            int s = 0;
            for (int i = 0; i < 128; ++i) { offs[i] = s; s += counts[i]; }
            offs[128] = s;
        }
        __syncthreads();
        int o = offs[t];
        int cbase = (chunk * 128 + t) * 32;
        while (wbits) {
            int bpos = __ffs(wbits) - 1;
            wbits &= wbits - 1u;
            nbr[o++] = (unsigned short)(cbase + bpos);
        }
        __syncthreads();
        int total = offs[128];
        for (int j = 0; j < total; ++j) {
            acc += xs[(int)nbr[j] * DIN + t];
        }
        __syncthreads();
    }
    h[(size_t)r * DIN + t] = dis[r] * acc;
}

// ---------------- 7. linear: out = h @ W^T + b via V_WMMA_F32_16X16X4_F32 ---
// Wave owns a 16x16 tile: 16 node rows x 16 output dims, K=128 -> 32 WMMAs.
// A (16x4 f32): lane m = lane&15; VGPR pair holds K {kk+kh, kk+kh+1},
//   kh = (lane>>4)*2.  B (4x16): lane n = lane&15, same K striping; B[k][n] =
//   W[n][k] so both operands read row (lane&15) of their source.
// C/D (16x16 f32): VGPR v -> M = v + (lane>=16 ? 8 : 0), N = lane&15.
__global__ __launch_bounds__(256) void k_gemm_wmma(const float* __restrict__ h,
                                                   const float* __restrict__ W,
                                                   const float* __restrict__ bias,
                                                   float* __restrict__ out) {
    int wave = threadIdx.x >> 5;            // 0..7 -> output-dim tile
    int lane = threadIdx.x & 31;
    int tileM = blockIdx.x;                 // 0..1023 (16384/16)
    int m16 = lane & 15;
    int kh = (lane >> 4) * 2;               // 0 or 2

    const float* ha = h + ((size_t)(tileM * 16 + m16)) * DIN;   // A row
    const float* wb = W + ((size_t)(wave * 16 + m16)) * DIN;    // W row = B col n

    v8f c = {0.f, 0.f, 0.f, 0.f, 0.f, 0.f, 0.f, 0.f};
    #pragma unroll
    for (int kk = 0; kk < DIN; kk += 4) {
        v2f a, b;
        a.x = ha[kk + kh]; a.y = ha[kk + kh + 1];
        b.x = wb[kk + kh]; b.y = wb[kk + kh + 1];
        c = __builtin_amdgcn_wmma_f32_16x16x4_f32(
                /*neg_a=*/false, a, /*neg_b=*/false, b,
                /*c_mod=*/(short)0, c, /*reuse_a=*/false, /*reuse_b=*/false);
    }

    int n = wave * 16 + m16;
    float bv = bias[n];
    int mBase = tileM * 16 + ((lane >> 4) ? 8 : 0);
    #pragma unroll
    for (int v = 0; v < 8; ++v) {
        out[(size_t)(mBase + v) * DOUT + n] = c[v] + bv;
    }
}

extern "C" void kernel_launch(void* const* d_in, const int* in_sizes, int n_in,
                              void* d_out, int out_size, void* d_ws, size_t ws_size,
                              hipStream_t stream) {
    const float* x    = (const float*)d_in[0];   // [16384,128]
    const int*   ei   = (const int*)d_in[1];     // [2,524288]
    const float* W    = (const float*)d_in[2];   // [128,128]
    const float* bias = (const float*)d_in[3];   // [128]
    float* out = (float*)d_out;                  // [16384,128]

    char* ws = (char*)d_ws;
    uint32_t* mask = (uint32_t*)ws;                                   // 32 MB
    float* dis = (float*)(ws + MASK_BYTES);                           // 64 KB
    float* xs  = (float*)(ws + MASK_BYTES + DIS_BYTES);               // 8 MB
    float* h   = (float*)(ws + MASK_BYTES + DIS_BYTES + XS_BYTES);    // 8 MB

    (void)in_sizes; (void)n_in; (void)out_size; (void)ws_size;

    k_zero_mask<<<MASK_WORDS / 4 / 256, 256, 0, stream>>>((uint4*)mask);
    k_diag     <<<NN / 256, 256, 0, stream>>>(mask);
    k_edges    <<<NEDGES / 256, 256, 0, stream>>>(ei, mask);
    k_degree   <<<NN, 128, 0, stream>>>(mask, dis);
    k_scale_x  <<<(NN * 32) / 256, 256, 0, stream>>>((const float4*)x, dis, (float4*)xs);
    k_aggregate<<<NN, 128, 0, stream>>>(mask, xs, dis, h);
    k_gemm_wmma<<<NN / 16, 256, 0, stream>>>(h, W, bias, out);
}